// GCN_79053168050931
// MI455X (gfx1250) — compile-verified
//
#include <hip/hip_runtime.h>
#include <math.h>

typedef __attribute__((ext_vector_type(2))) float v2f;
typedef __attribute__((ext_vector_type(8))) float v8f;

#define THREADS 256

// ---------------- gcn_norm precompute ----------------
__global__ void k_deg_init(float* __restrict__ deg, int n) {
  int i = blockIdx.x * blockDim.x + threadIdx.x;
  if (i < n) deg[i] = 1.0f;  // self-loop weight
}

__global__ void k_deg_acc(float* __restrict__ deg, const int* __restrict__ dst,
                          const float* __restrict__ ew, int e) {
  int i = blockIdx.x * blockDim.x + threadIdx.x;
  if (i < e) atomicAdd(&deg[dst[i]], ew[i]);
}

__global__ void k_dinv(float* __restrict__ deg, int n) {
  int i = blockIdx.x * blockDim.x + threadIdx.x;
  if (i < n) {
    float d = deg[i];
    deg[i] = (d > 0.0f) ? rsqrtf(d) : 0.0f;  // deg -> dinv in place
  }
}

__global__ void k_norm(float* __restrict__ norm, const int* __restrict__ src,
                       const int* __restrict__ dst, const float* __restrict__ ew,
                       const float* __restrict__ dinv, int e) {
  int i = blockIdx.x * blockDim.x + threadIdx.x;
  if (i < e) norm[i] = dinv[src[i]] * ew[i] * dinv[dst[i]];
}

// ---------------- WMMA GEMM: T[n, FOUT] = X[n, 128] @ W[128, FOUT] ----------------
// One wave computes a 16-row strip across all FOUT columns.
// A fragments (16x4 f32) held in registers across column tiles; W staged in LDS.
template <int FOUT>
__global__ __launch_bounds__(THREADS) void k_gemm(const float* __restrict__ X,
                                                  const float* __restrict__ W,
                                                  float* __restrict__ T, int nrows) {
  constexpr int LSTR = FOUT + 16;  // pad so (k+2)*LSTR lands on different banks
  __shared__ float Wlds[128 * LSTR];
  for (int i = threadIdx.x; i < 128 * FOUT; i += THREADS) {
    int k = i / FOUT, c = i % FOUT;
    Wlds[k * LSTR + c] = W[i];
  }
  __syncthreads();

  const int wave  = threadIdx.x >> 5;
  const int lane  = threadIdx.x & 31;
  const int m     = lane & 15;   // M (for A) / N (for B,D)
  const int khalf = lane >> 4;   // 0: K+0/1, 1: K+2/3
  const int tile  = blockIdx.x * (THREADS / 32) + wave;
  if (tile * 16 >= nrows) return;  // wave-uniform: EXEC stays all-1s for WMMA
  const int row0 = tile * 16;

  // A: 16x4 f32 layout -> lane m holds {A[m][k0+2*khalf], A[m][k0+2*khalf+1]}
  v2f afrag[32];
  const float* xrow = X + (size_t)(row0 + m) * 128 + 2 * khalf;
#pragma unroll
  for (int kk = 0; kk < 32; ++kk) {
    afrag[kk].x = xrow[kk * 4 + 0];
    afrag[kk].y = xrow[kk * 4 + 1];
  }

#pragma unroll
  for (int ct = 0; ct < FOUT / 16; ++ct) {
    v8f acc = {};
    const float* wb = &Wlds[ct * 16 + m];  // column n = m
#pragma unroll
    for (int kk = 0; kk < 32; ++kk) {
      const int k = kk * 4 + 2 * khalf;
      v2f b;  // B: 4x16 f32 -> lane m holds {B[k][n], B[k+1][n]}
      b.x = wb[(k + 0) * LSTR];
      b.y = wb[(k + 1) * LSTR];
      acc = __builtin_amdgcn_wmma_f32_16x16x4_f32(
          /*neg_a=*/false, afrag[kk], /*neg_b=*/false, b,
          /*c_mod=*/(short)0, acc, /*reuse_a=*/false, /*reuse_b=*/false);
    }
    // D layout: VGPR r -> M = r + 8*khalf, N = m
#pragma unroll
    for (int r = 0; r < 8; ++r)
      T[(size_t)(row0 + r + 8 * khalf) * FOUT + ct * 16 + m] = acc[r];
  }
}

// ---------------- aggregation ----------------
// out[i,:] = T[i,:] * dinv[i]^2   (self-loop term; also initializes the buffer)
template <int FOUT>
__global__ void k_selfloop_init(const float* __restrict__ T, const float* __restrict__ dinv,
                                float* __restrict__ out, int n) {
  int idx = blockIdx.x * blockDim.x + threadIdx.x;  // over n * FOUT/4
  if (idx >= n * (FOUT / 4)) return;
  int i = idx / (FOUT / 4);
  int f = (idx % (FOUT / 4)) * 4;
  float s = dinv[i];
  s = s * s;
  const float4 v = *(const float4*)(T + (size_t)i * FOUT + f);
  float4 o;
  o.x = v.x * s; o.y = v.y * s; o.z = v.z * s; o.w = v.w * s;
  *(float4*)(out + (size_t)i * FOUT + f) = o;
}

// out[dst[e], :] += T[src[e], :] * norm[e]  — FOUT/4 threads per edge, float4 each
template <int FOUT>
__global__ void k_scatter(const float* __restrict__ T, const int* __restrict__ src,
                          const int* __restrict__ dst, const float* __restrict__ norm,
                          float* __restrict__ out, int e) {
  long long idx = (long long)blockIdx.x * blockDim.x + threadIdx.x;
  long long total = (long long)e * (FOUT / 4);
  if (idx >= total) return;
  int ed = (int)(idx / (FOUT / 4));
  int f = (int)(idx % (FOUT / 4)) * 4;
  float nw = norm[ed];
  const float4 v = *(const float4*)(T + (size_t)src[ed] * FOUT + f);
  float* o = out + (size_t)dst[ed] * FOUT + f;
  atomicAdd(o + 0, v.x * nw);
  atomicAdd(o + 1, v.y * nw);
  atomicAdd(o + 2, v.z * nw);
  atomicAdd(o + 3, v.w * nw);
}

template <int FOUT>
__global__ void k_bias_relu(float* __restrict__ h, const float* __restrict__ b, int n) {
  int idx = blockIdx.x * blockDim.x + threadIdx.x;  // over n * FOUT
  if (idx >= n * FOUT) return;
  int f = idx % FOUT;
  h[idx] = fmaxf(h[idx] + b[f], 0.0f);
}

// ---------------- final: bias + relu + log_softmax over 64 cols, in place ----------------
__global__ void k_bias_relu_logsoftmax64(float* __restrict__ out, const float* __restrict__ b,
                                         int n) {
  int wave = threadIdx.x >> 5;
  int lane = threadIdx.x & 31;
  int row = blockIdx.x * (THREADS / 32) + wave;
  if (row >= n) return;
  float* o = out + (size_t)row * 64;
  float v0 = fmaxf(o[lane] + b[lane], 0.0f);
  float v1 = fmaxf(o[lane + 32] + b[lane + 32], 0.0f);
  float m = fmaxf(v0, v1);
#pragma unroll
  for (int off = 16; off > 0; off >>= 1) m = fmaxf(m, __shfl_xor(m, off, 32));
  float s = expf(v0 - m) + expf(v1 - m);
#pragma unroll
  for (int off = 16; off > 0; off >>= 1) s += __shfl_xor(s, off, 32);
  float lse = logf(s) + m;
  o[lane] = v0 - lse;
  o[lane + 32] = v1 - lse;
}

// ---------------- host launcher ----------------
static inline int cdiv(long long a, long long b) { return (int)((a + b - 1) / b); }

extern "C" void kernel_launch(void* const* d_in, const int* in_sizes, int n_in,
                              void* d_out, int out_size, void* d_ws, size_t ws_size,
                              hipStream_t stream) {
  const float* x   = (const float*)d_in[0];
  const int*   src = (const int*)d_in[1];
  const float* ew  = (const float*)d_in[2];
  const float* W1  = (const float*)d_in[3];
  const float* b1  = (const float*)d_in[4];
  const float* W2  = (const float*)d_in[5];
  const float* b2  = (const float*)d_in[6];
  const float* W3  = (const float*)d_in[7];
  const float* b3  = (const float*)d_in[8];
  float* out = (float*)d_out;

  const int N = in_sizes[0] / 128;
  const int E = in_sizes[1] / 2;
  const int* dst = src + E;

  // workspace layout (floats), float4-aligned sections
  float* dinv = (float*)d_ws;            // N   (deg, then dinv in place)
  float* norm = dinv + N;                // E
  float* t    = norm + E;                // N*128 (gemm output)
  float* h    = t + (size_t)N * 128;     // N*128 (aggregated layer output)

  // --- gcn_norm (cached across layers) ---
  k_deg_init<<<cdiv(N, THREADS), THREADS, 0, stream>>>(dinv, N);
  k_deg_acc<<<cdiv(E, THREADS), THREADS, 0, stream>>>(dinv, dst, ew, E);
  k_dinv<<<cdiv(N, THREADS), THREADS, 0, stream>>>(dinv, N);
  k_norm<<<cdiv(E, THREADS), THREADS, 0, stream>>>(norm, src, dst, ew, dinv, E);

  const int gemm_blocks = cdiv((long long)cdiv(N, 16), THREADS / 32);

  // --- layer 1: h = relu(Agg(x @ W1) + b1) ---
  k_gemm<128><<<gemm_blocks, THREADS, 0, stream>>>(x, W1, t, N);
  k_selfloop_init<128><<<cdiv((long long)N * 32, THREADS), THREADS, 0, stream>>>(t, dinv, h, N);
  k_scatter<128><<<cdiv((long long)E * 32, THREADS), THREADS, 0, stream>>>(t, src, dst, norm, h, E);
  k_bias_relu<128><<<cdiv((long long)N * 128, THREADS), THREADS, 0, stream>>>(h, b1, N);

  // --- layer 2: h = relu(Agg(h @ W2) + b2) ---
  k_gemm<128><<<gemm_blocks, THREADS, 0, stream>>>(h, W2, t, N);
  k_selfloop_init<128><<<cdiv((long long)N * 32, THREADS), THREADS, 0, stream>>>(t, dinv, h, N);
  k_scatter<128><<<cdiv((long long)E * 32, THREADS), THREADS, 0, stream>>>(t, src, dst, norm, h, E);
  k_bias_relu<128><<<cdiv((long long)N * 128, THREADS), THREADS, 0, stream>>>(h, b2, N);

  // --- layer 3 (FOUT=64): aggregate straight into d_out, then log_softmax in place ---
  k_gemm<64><<<gemm_blocks, THREADS, 0, stream>>>(h, W3, t, N);
  k_selfloop_init<64><<<cdiv((long long)N * 16, THREADS), THREADS, 0, stream>>>(t, dinv, out, N);
  k_scatter<64><<<cdiv((long long)E * 16, THREADS), THREADS, 0, stream>>>(t, src, dst, norm, out, E);
  k_bias_relu_logsoftmax64<<<cdiv(N, THREADS / 32), THREADS, 0, stream>>>(out, b3, N);
}